// VNG_38783554683426
// MI455X (gfx1250) — compile-verified
//
#include <hip/hip_runtime.h>
#include <hip/hip_bf16.h>

#define ALPHA 0.1f
#define CDIM 32
#define NITER 10

typedef float v2f __attribute__((ext_vector_type(2)));
typedef float v8f __attribute__((ext_vector_type(8)));

// ---------------------------------------------------------------------------
// preds0 = pi_mat^T   (pi_mat is [C, N] row-major -> preds0 [N, C])
// LDS-tiled 32x32 transpose: coalesced on both sides.
// ---------------------------------------------------------------------------
__global__ void VNG_transpose_pi(const float* __restrict__ pi,
                                 float* __restrict__ preds0, int N) {
    __shared__ float tile[32][33];
    const int tx = threadIdx.x;      // 0..31
    const int ty = threadIdx.y;      // 0..7
    const int n0 = blockIdx.x * 32;

    for (int yy = ty; yy < 32; yy += 8) {
        int n = n0 + tx;
        tile[yy][tx] = (n < N) ? pi[(size_t)yy * N + n] : 0.0f;
    }
    __syncthreads();
    for (int yy = ty; yy < 32; yy += 8) {
        int n = n0 + yy;
        if (n < N) preds0[(size_t)n * CDIM + tx] = tile[tx][yy];
    }
}

// ---------------------------------------------------------------------------
// res = ALPHA * local_preds, once, via V_WMMA_F32_16X16X4_F32 (exact in f32).
// D = sum_j A_j x B_j, A_j[m][k] = ALPHA*delta(m,4j+k), B_j = rows 4j..4j+3.
// Layouts (ISA 7.12.2, 32-bit, wave32):
//   A(16x4): lane L, vgpr p -> A[L&15][p + 2*(L>>4)]
//   B(4x16): lane L, vgpr p -> B[p + 2*(L>>4)][L&15]
//   C/D    : lane L, vgpr p -> D[p + 8*(L>>4)][L&15]
// ---------------------------------------------------------------------------
__global__ void VNG_residual_wmma(const float* __restrict__ local_preds,
                                  float* __restrict__ res, int N) {
    const int lane = threadIdx.x & 31;
    const int wave = (blockIdx.x * blockDim.x + threadIdx.x) >> 5;
    const int rowTiles = (N + 15) >> 4;
    const int nTiles = rowTiles * (CDIM / 16);
    if (wave >= nTiles) return;          // wave-uniform: EXEC stays all-1s

    const int tr = wave >> 1;
    const int tc = wave & 1;
    const int row0 = tr * 16;
    const int col0 = tc * 16;
    const int hi = lane >> 4;
    const int lo = lane & 15;

    v8f acc = {0.f, 0.f, 0.f, 0.f, 0.f, 0.f, 0.f, 0.f};

    #pragma unroll
    for (int j = 0; j < 4; ++j) {
        int r0 = row0 + 4 * j + 0 + 2 * hi; r0 = r0 < N ? r0 : N - 1;
        int r1 = row0 + 4 * j + 1 + 2 * hi; r1 = r1 < N ? r1 : N - 1;
        v2f b;
        b.x = local_preds[(size_t)r0 * CDIM + col0 + lo];
        b.y = local_preds[(size_t)r1 * CDIM + col0 + lo];
        v2f a;
        a.x = (lo == (4 * j + 0 + 2 * hi)) ? ALPHA : 0.0f;
        a.y = (lo == (4 * j + 1 + 2 * hi)) ? ALPHA : 0.0f;
        acc = __builtin_amdgcn_wmma_f32_16x16x4_f32(
            false, a, false, b, (short)0, acc, false, false);
    }

    if (row0 + 16 <= N) {                 // wave-uniform fast path: no guards
        #pragma unroll
        for (int p = 0; p < 8; ++p)
            res[(size_t)(row0 + p + 8 * hi) * CDIM + col0 + lo] = acc[p];
    } else {
        #pragma unroll
        for (int p = 0; p < 8; ++p) {
            int m = row0 + p + 8 * hi;
            if (m < N) res[(size_t)m * CDIM + col0 + lo] = acc[p];
        }
    }
}

// ---------------------------------------------------------------------------
// CSR construction (edge list is static across all NITER iterations).
// ---------------------------------------------------------------------------
__global__ void VNG_zero_i32(int* __restrict__ p, int n) {
    int i = blockIdx.x * blockDim.x + threadIdx.x;
    if (i < n) p[i] = 0;
}

__global__ void VNG_degree_hist(const int* __restrict__ rows,
                                int* __restrict__ deg, int E) {
    int e = blockIdx.x * blockDim.x + threadIdx.x;
    if (e < E) atomicAdd(&deg[rows[e]], 1);
}

// Single-block exclusive scan: chunked Hillis-Steele with inter-chunk carry.
// Reads deg[] (== cursor array) in place, writes offs[0..N] and cursor[0..N-1].
__global__ void VNG_scan_offsets(int* __restrict__ cursor /* in: deg */,
                                 int* __restrict__ offs, int N) {
    __shared__ int sbuf[1024];
    __shared__ int carry_s;
    if (threadIdx.x == 0) carry_s = 0;
    __syncthreads();
    for (int base = 0; base < N; base += 1024) {
        int i = base + (int)threadIdx.x;
        int v = (i < N) ? cursor[i] : 0;
        sbuf[threadIdx.x] = v;
        __syncthreads();
        for (int off = 1; off < 1024; off <<= 1) {
            int t = ((int)threadIdx.x >= off) ? sbuf[threadIdx.x - off] : 0;
            __syncthreads();
            sbuf[threadIdx.x] += t;
            __syncthreads();
        }
        int carry = carry_s;
        int excl = carry + sbuf[threadIdx.x] - v;
        if (i < N) { offs[i] = excl; cursor[i] = excl; }
        __syncthreads();
        if (threadIdx.x == 1023) carry_s = carry + sbuf[1023];
        __syncthreads();
    }
    if (threadIdx.x == 0) offs[N] = carry_s;
}

__global__ void VNG_csr_fill(const int* __restrict__ rows,
                             const int* __restrict__ cols,
                             const float* __restrict__ vals,
                             int* __restrict__ cursor,
                             int* __restrict__ ccol,
                             float* __restrict__ cval, int E) {
    int e = blockIdx.x * blockDim.x + threadIdx.x;
    if (e >= E) return;
    int r = rows[e];
    int pos = atomicAdd(&cursor[r], 1);   // int atomics only, build-time only
    ccol[pos] = cols[e];
    cval[pos] = vals[e];
}

// ---------------------------------------------------------------------------
// Pull-based SpMM row body: acc = res[row,:] + sum_j cval[j]*cur[ccol[j],:]
// lane = feature (C == wave32 width). Broadcast loads of col/val; coalesced
// 128B row gathers, L2-resident; 4-wide unroll keeps 4 gathers in flight.
// ---------------------------------------------------------------------------
__device__ __forceinline__ float VNG_row_accum(const int* __restrict__ ccol,
                                               const float* __restrict__ cval,
                                               const float* __restrict__ cur,
                                               int beg, int end, int lane,
                                               float acc) {
    int j = beg;
    for (; j + 3 < end; j += 4) {
        int   c0 = ccol[j],     c1 = ccol[j + 1];
        int   c2 = ccol[j + 2], c3 = ccol[j + 3];
        float v0 = cval[j],     v1 = cval[j + 1];
        float v2 = cval[j + 2], v3 = cval[j + 3];
        float x0 = cur[(size_t)c0 * CDIM + lane];
        float x1 = cur[(size_t)c1 * CDIM + lane];
        float x2 = cur[(size_t)c2 * CDIM + lane];
        float x3 = cur[(size_t)c3 * CDIM + lane];
        acc = fmaf(v0, x0, acc);
        acc = fmaf(v1, x1, acc);
        acc = fmaf(v2, x2, acc);
        acc = fmaf(v3, x3, acc);
    }
    for (; j < end; ++j) {
        int   c = ccol[j];
        float v = cval[j];
        acc = fmaf(v, cur[(size_t)c * CDIM + lane], acc);
    }
    return acc;
}

// Full sweep: one wave per node row, writes nxt[N, C].
__global__ void VNG_spmv_pull(const int* __restrict__ offs,
                              const int* __restrict__ ccol,
                              const float* __restrict__ cval,
                              const float* __restrict__ cur,
                              const float* __restrict__ res,
                              float* __restrict__ nxt, int N) {
    const int lane = threadIdx.x & 31;
    const int row = (blockIdx.x * blockDim.x + threadIdx.x) >> 5;
    if (row >= N) return;

    float acc = res[(size_t)row * CDIM + lane];
    acc = VNG_row_accum(ccol, cval, cur, offs[row], offs[row + 1], lane, acc);
    nxt[(size_t)row * CDIM + lane] = acc;
}

// Final sweep fused with output gather: only the NIDX indexed rows are
// computed (10x less work than a full sweep), written straight to d_out.
__global__ void VNG_spmv_pull_idx(const int* __restrict__ offs,
                                  const int* __restrict__ ccol,
                                  const float* __restrict__ cval,
                                  const float* __restrict__ cur,
                                  const float* __restrict__ res,
                                  const int* __restrict__ idx,
                                  float* __restrict__ out, int nidx) {
    const int lane = threadIdx.x & 31;
    const int i = (blockIdx.x * blockDim.x + threadIdx.x) >> 5;
    if (i >= nidx) return;

    const int row = idx[i];                       // wave-uniform broadcast
    float acc = res[(size_t)row * CDIM + lane];
    acc = VNG_row_accum(ccol, cval, cur, offs[row], offs[row + 1], lane, acc);
    out[(size_t)i * CDIM + lane] = acc;
}

// ---------------------------------------------------------------------------
extern "C" void kernel_launch(void* const* d_in, const int* in_sizes, int n_in,
                              void* d_out, int out_size, void* d_ws, size_t ws_size,
                              hipStream_t stream) {
    const int*   edge_rows   = (const int*)  d_in[0];
    const int*   edge_cols   = (const int*)  d_in[1];
    const float* edge_vals   = (const float*)d_in[2];
    const float* pi_mat      = (const float*)d_in[3];
    const float* local_preds = (const float*)d_in[4];
    const int*   idx         = (const int*)  d_in[5];

    const int E    = in_sizes[0];
    const int N    = in_sizes[4] / CDIM;
    const int NIDX = in_sizes[5];
    const size_t NC = (size_t)N * CDIM;

    // Workspace layout (all 4-byte elements): ~52 MB for N=100K, E=1.6M
    float* buf0   = (float*)d_ws;            // ping  [N*C]
    float* buf1   = buf0 + NC;               // pong  [N*C]
    float* res    = buf1 + NC;               // alpha*local_preds [N*C]
    float* cval   = res + NC;                // CSR values [E]
    int*   ccol   = (int*)(cval + E);        // CSR cols   [E]
    int*   offs   = ccol + E;                // row offsets [N+1]
    int*   cursor = offs + (N + 1);          // degree/cursor [N]

    const int eBlocks = (E + 255) / 256;
    const int nBlocks = (N + 255) / 256;

    // 1. preds0 = pi_mat^T
    VNG_transpose_pi<<<(N + 31) / 32, dim3(32, 8), 0, stream>>>(pi_mat, buf0, N);

    // 2. CSR build (once; reused for all 10 iterations)
    VNG_zero_i32    <<<nBlocks, 256, 0, stream>>>(cursor, N);
    VNG_degree_hist <<<eBlocks, 256, 0, stream>>>(edge_rows, cursor, E);
    VNG_scan_offsets<<<1, 1024, 0, stream>>>(cursor, offs, N);
    VNG_csr_fill    <<<eBlocks, 256, 0, stream>>>(edge_rows, edge_cols, edge_vals,
                                                  cursor, ccol, cval, E);

    // 3. res = ALPHA * local_preds, once, via WMMA
    const int nTiles = ((N + 15) >> 4) * (CDIM / 16);
    VNG_residual_wmma<<<(nTiles * 32 + 255) / 256, 256, 0, stream>>>(
        local_preds, res, N);

    // 4. NITER-1 full pull sweeps (no float atomics)
    const int pullBlocks = (int)(((size_t)N * 32 + 255) / 256);
    float* cur = buf0;
    float* nxt = buf1;
    for (int it = 0; it < NITER - 1; ++it) {
        VNG_spmv_pull<<<pullBlocks, 256, 0, stream>>>(offs, ccol, cval,
                                                      cur, res, nxt, N);
        float* tmp = cur; cur = nxt; nxt = tmp;
    }

    // 5. Final sweep fused with the idx gather: compute only NIDX rows.
    VNG_spmv_pull_idx<<<(NIDX * 32 + 255) / 256, 256, 0, stream>>>(
        offs, ccol, cval, cur, res, idx, (float*)d_out, NIDX);
}